// StabilizeNet_85246510891697
// MI455X (gfx1250) — compile-verified
//
#include <hip/hip_runtime.h>
#include <hip/hip_bf16.h>
#include <math.h>

// Problem constants (match reference)
#define TT 512
#define BB 256
#define XX 128
#define HH 512
#define OO 128

typedef float v2f __attribute__((ext_vector_type(2)));
typedef float v8f __attribute__((ext_vector_type(8)));

#define KC  32   // K-chunk staged in LDS
#define LDP 36   // padded LDS row stride: 16B-aligned rows (b128 transfers),
                 // 36r mod 64 = 4*(9r mod 16) -> conflict-free 16-row frag reads

__device__ __forceinline__ v8f wmma_f32(v2f a, v2f b, v8f c) {
    // D = A(16x4) * B(4x16) + C, fp32
    return __builtin_amdgcn_wmma_f32_16x16x4_f32(false, a, false, b, (short)0, c, false, false);
}

__device__ __forceinline__ float sigmoidf(float x) { return 1.0f / (1.0f + expf(-x)); }

// CDNA5 async global->LDS (ASYNCcnt-tracked, no VGPR staging)
__device__ __forceinline__ void async_ld_b128(void* lds_dst, const void* gsrc) {
    asm volatile("global_load_async_to_lds_b128 %0, %1, off"
                 :: "v"((__attribute__((address_space(3))) void*)lds_dst),
                    "v"(gsrc)
                 : "memory");
}
__device__ __forceinline__ void wait_async0() {
    asm volatile("s_wait_asynccnt 0x0" ::: "memory");
}

// ---------------------------------------------------------------------------
// Generic fp32 WMMA GEMM:  C[M,N] = A[M,K] @ W[N,K]^T  (+ bias / relu epilogue)
// block = 256 threads (8 waves), tile 64x64, async double-buffered K chunks.
// mode: 0 = plain, 1 = relu(.+bias), 2 = .+bias
// ---------------------------------------------------------------------------
__global__ __launch_bounds__(256) void gemm_f32_wmma(
    const float* __restrict__ A, const float* __restrict__ W,
    float* __restrict__ C, const float* __restrict__ bias,
    int M, int N, int K, int mode)
{
    __shared__ __align__(16) float As[2][64][LDP];
    __shared__ __align__(16) float Bs[2][64][LDP];

    const int tid  = threadIdx.x;
    const int lane = tid & 31;
    const int wave = tid >> 5;
    const int tm = blockIdx.y * 64;
    const int tn = blockIdx.x * 64;
    const int sm = (wave & 3) * 16;        // wave's row sub-tile
    const int sn = (wave >> 2) * 32;       // wave's 2 col sub-tiles (sn, sn+16)

    const int lrow = tid >> 3;             // 0..31 (32 rows per pass)
    const int lcol = (tid & 7) * 4;        // 0..28 step 4

    const int half = lane >> 4;            // 0|1
    const int lr   = lane & 15;

    v8f acc0 = {}; v8f acc1 = {};

    auto issue_chunk = [&](int buf, int k0) {
        #pragma unroll
        for (int p = 0; p < 2; ++p) {
            const int r = lrow + p * 32;
            async_ld_b128(&As[buf][r][lcol], A + (size_t)(tm + r) * K + k0 + lcol);
            async_ld_b128(&Bs[buf][r][lcol], W + (size_t)(tn + r) * K + k0 + lcol);
        }
    };

    const int nch = K / KC;
    issue_chunk(0, 0);

    for (int c = 0; c < nch; ++c) {
        const int cur = c & 1;
        wait_async0();                         // own async ops for chunk c done
        __syncthreads();                       // all waves' data visible
        if (c + 1 < nch) issue_chunk(cur ^ 1, (c + 1) * KC);   // prefetch next

        #pragma unroll
        for (int kk = 0; kk < KC; kk += 4) {
            const int kf = kk + 2 * half;  // K = vgpr + 2*half
            v2f a, b0, b1;
            a.x  = As[cur][sm + lr][kf];        a.y  = As[cur][sm + lr][kf + 1];
            b0.x = Bs[cur][sn + lr][kf];        b0.y = Bs[cur][sn + lr][kf + 1];
            b1.x = Bs[cur][sn + 16 + lr][kf];   b1.y = Bs[cur][sn + 16 + lr][kf + 1];
            acc0 = wmma_f32(a, b0, acc0);
            acc1 = wmma_f32(a, b1, acc1);
        }
    }

    // epilogue: C/D layout -> M = v + 8*half, N = lr
    #pragma unroll
    for (int v = 0; v < 8; ++v) {
        const int m  = tm + sm + v + 8 * half;
        const int n0 = tn + sn + lr;
        const int n1 = n0 + 16;
        float c0 = acc0[v], c1 = acc1[v];
        if (mode != 0) {
            c0 += bias[n0]; c1 += bias[n1];
            if (mode == 1) { c0 = fmaxf(c0, 0.0f); c1 = fmaxf(c1, 0.0f); }
        }
        C[(size_t)m * N + n0] = c0;
        C[(size_t)m * N + n1] = c1;
    }
}

// ---------------------------------------------------------------------------
// Fused GRU step: gates = h_prev @ W_hh^T + x_t @ W_ih^T, then r/z/n math.
// block = 128 threads (4 waves), tile 32(batch) x 32(hidden cols).
// Async double-buffered, unified 20-chunk K loop: chunks 0..15 = hidden
// (K=512), chunks 16..19 = input (K=128). N-gate split into input/hidden
// accumulators because n = tanh(i_n + r * h_n).
// ---------------------------------------------------------------------------
#define NCHH (HH / KC)            // 16 hidden chunks
#define NCH  ((HH + XX) / KC)     // 20 total chunks

__global__ __launch_bounds__(128) void gru_step(
    const float* __restrict__ x_t,     // [B, X]
    const float* __restrict__ h_prev,  // [B, H] (zero buffer when t==0)
    const float* __restrict__ W_ih,    // [3H, X]
    const float* __restrict__ W_hh,    // [3H, H]
    float* __restrict__ h_out)         // [B, H]
{
    __shared__ __align__(16) float As[2][32][LDP];
    __shared__ __align__(16) float Bs[2][3][32][LDP];

    const int tid  = threadIdx.x;
    const int lane = tid & 31;
    const int wave = tid >> 5;
    const int tm = blockIdx.y * 32;        // batch tile
    const int tn = blockIdx.x * 32;        // hidden-col tile
    const int sm = (wave & 1) * 16;
    const int sn = (wave >> 1) * 16;

    const int lrow = tid >> 3;             // 0..15 (16 rows per pass)
    const int lcol = (tid & 7) * 4;
    const int half = lane >> 4;
    const int lr   = lane & 15;

    v8f aR = {}; v8f aZ = {}; v8f aNi = {}; v8f aNh = {};

    auto issue_chunk = [&](int buf, int c) {
        const bool hpart = (c < NCHH);
        const int k0       = (hpart ? c : c - NCHH) * KC;
        const float* Asrc  = hpart ? h_prev : x_t;
        const float* Wsrc  = hpart ? W_hh : W_ih;
        const int ldk      = hpart ? HH : XX;
        #pragma unroll
        for (int p = 0; p < 2; ++p) {
            const int r = lrow + p * 16;
            async_ld_b128(&As[buf][r][lcol],
                          Asrc + (size_t)(tm + r) * ldk + k0 + lcol);
            #pragma unroll
            for (int g = 0; g < 3; ++g)
                async_ld_b128(&Bs[buf][g][r][lcol],
                              Wsrc + (size_t)(g * HH + tn + r) * ldk + k0 + lcol);
        }
    };

    issue_chunk(0, 0);

    for (int c = 0; c < NCH; ++c) {
        const int cur = c & 1;
        wait_async0();
        __syncthreads();
        if (c + 1 < NCH) issue_chunk(cur ^ 1, c + 1);   // prefetch next chunk

        const bool hpart = (c < NCHH);                  // uniform branch
        #pragma unroll
        for (int kk = 0; kk < KC; kk += 4) {
            const int kf = kk + 2 * half;
            v2f a, bR, bZ, bN;
            a.x  = As[cur][sm + lr][kf];        a.y  = As[cur][sm + lr][kf + 1];
            bR.x = Bs[cur][0][sn + lr][kf];     bR.y = Bs[cur][0][sn + lr][kf + 1];
            bZ.x = Bs[cur][1][sn + lr][kf];     bZ.y = Bs[cur][1][sn + lr][kf + 1];
            bN.x = Bs[cur][2][sn + lr][kf];     bN.y = Bs[cur][2][sn + lr][kf + 1];
            aR = wmma_f32(a, bR, aR);
            aZ = wmma_f32(a, bZ, aZ);
            if (hpart) aNh = wmma_f32(a, bN, aNh);
            else       aNi = wmma_f32(a, bN, aNi);
        }
    }

    // ---- gate math + store h_new ----
    #pragma unroll
    for (int v = 0; v < 8; ++v) {
        const int m = tm + sm + v + 8 * half;
        const int j = tn + sn + lr;
        const float r  = sigmoidf(aR[v]);
        const float z  = sigmoidf(aZ[v]);
        const float nn = tanhf(aNi[v] + r * aNh[v]);
        const float hp = h_prev[(size_t)m * HH + j];
        h_out[(size_t)m * HH + j] = (1.0f - z) * nn + z * hp;
    }
}

// ---------------------------------------------------------------------------
// Deterministic two-stage sum((preds-gt)^2)
// ---------------------------------------------------------------------------
#define NRED 4096
__global__ __launch_bounds__(256) void sq_partial(
    const float* __restrict__ preds, const float* __restrict__ gt,
    float* __restrict__ part, int n)
{
    __shared__ float s[256];
    float acc = 0.0f;
    for (int i = blockIdx.x * 256 + threadIdx.x; i < n; i += gridDim.x * 256) {
        const float d = preds[i] - gt[i];
        acc += d * d;
    }
    s[threadIdx.x] = acc;
    __syncthreads();
    for (int st = 128; st > 0; st >>= 1) {
        if (threadIdx.x < st) s[threadIdx.x] += s[threadIdx.x + st];
        __syncthreads();
    }
    if (threadIdx.x == 0) part[blockIdx.x] = s[0];
}

__global__ __launch_bounds__(256) void final_reduce(
    const float* __restrict__ part, int n, float* __restrict__ out)
{
    __shared__ float s[256];
    float acc = 0.0f;
    for (int i = threadIdx.x; i < n; i += 256) acc += part[i];
    s[threadIdx.x] = acc;
    __syncthreads();
    for (int st = 128; st > 0; st >>= 1) {
        if (threadIdx.x < st) s[threadIdx.x] += s[threadIdx.x + st];
        __syncthreads();
    }
    if (threadIdx.x == 0) out[0] = s[0];
}

// ---------------------------------------------------------------------------
extern "C" void kernel_launch(void* const* d_in, const int* in_sizes, int n_in,
                              void* d_out, int out_size, void* d_ws, size_t ws_size,
                              hipStream_t stream) {
    (void)in_sizes; (void)n_in; (void)out_size; (void)ws_size;
    const float* x    = (const float*)d_in[0];  // [T,B,X]
    const float* gt   = (const float*)d_in[1];  // [T,B,O]
    const float* W_ih = (const float*)d_in[2];  // [3H,X]
    const float* W_hh = (const float*)d_in[3];  // [3H,H]
    const float* W1   = (const float*)d_in[4];  // [H,H]
    const float* b1   = (const float*)d_in[5];  // [H]
    const float* W2   = (const float*)d_in[6];  // [O,H]
    const float* b2   = (const float*)d_in[7];  // [O]
    float* out = (float*)d_out;                 // [0]=loss, [1..]=preds [T,1,B,O]

    // workspace layout
    float* Hseq = (float*)d_ws;                       // T*B*H
    float* A1   = Hseq + (size_t)TT * BB * HH;        // T*B*H (decoder hidden)
    float* zbuf = A1   + (size_t)TT * BB * HH;        // B*H zero h0
    float* part = zbuf + (size_t)BB * HH;             // NRED partials

    hipMemsetAsync(zbuf, 0, (size_t)BB * HH * sizeof(float), stream);

    // sequential GRU: one fused WMMA kernel per timestep
    for (int t = 0; t < TT; ++t) {
        const float* hp = (t == 0) ? zbuf : (Hseq + (size_t)(t - 1) * BB * HH);
        gru_step<<<dim3(HH / 32, BB / 32), 128, 0, stream>>>(
            x + (size_t)t * BB * XX, hp, W_ih, W_hh, Hseq + (size_t)t * BB * HH);
    }

    // decoder over all timesteps at once (M = T*B)
    const int M = TT * BB;
    gemm_f32_wmma<<<dim3(HH / 64, M / 64), 256, 0, stream>>>(
        Hseq, W1, A1, b1, M, HH, HH, /*relu+bias*/ 1);
    gemm_f32_wmma<<<dim3(OO / 64, M / 64), 256, 0, stream>>>(
        A1, W2, out + 1, b2, M, OO, HH, /*bias*/ 2);

    // loss = sum((preds - gt)^2)
    sq_partial<<<NRED, 256, 0, stream>>>(out + 1, gt, part, M * OO);
    final_reduce<<<1, 256, 0, stream>>>(part, NRED, out);
}